// BiGAT_53996328845504
// MI455X (gfx1250) — compile-verified
//
#include <hip/hip_runtime.h>
#include <hip/hip_bf16.h>
#include <math.h>

typedef __attribute__((ext_vector_type(2))) float v2f;
typedef __attribute__((ext_vector_type(8))) float v8f;

#define NEG_SLOPE 0.2f

// ---- order-preserving float <-> uint (for atomic segment-max) ----
static __device__ __forceinline__ unsigned f2ord(float f) {
  unsigned u = __float_as_uint(f);
  return (u & 0x80000000u) ? ~u : (u | 0x80000000u);
}
static __device__ __forceinline__ float ord2f(unsigned u) {
  return (u & 0x80000000u) ? __uint_as_float(u & 0x7fffffffu)
                           : __uint_as_float(~u);
}
// encoded -inf: ~0xFF800000
#define ORD_NEG_INF 0x007FFFFFu

// ------------------------------ init kernels ------------------------------
__global__ void fill_f32_kernel(float* __restrict__ p, float v, int n) {
  int i = blockIdx.x * blockDim.x + threadIdx.x;
  if (i < n) p[i] = v;
}
__global__ void fill_u32_kernel(unsigned* __restrict__ p, unsigned v, int n) {
  int i = blockIdx.x * blockDim.x + threadIdx.x;
  if (i < n) p[i] = v;
}
__global__ void init_bias_kernel(float* __restrict__ out,
                                 const float* __restrict__ b, int n, int cmask) {
  int i = blockIdx.x * blockDim.x + threadIdx.x;
  if (i < n) out[i] = b[i & cmask];
}

// ------------------------------ fp32 WMMA GEMM ----------------------------
// Y[M x NC] = X[M x K] * W[K x NC], all row-major.
// One wave computes one 16x16 tile with V_WMMA_F32_16X16X4_F32 (K/4 steps).
// blockIdx.x = row tile (M/16 blocks, M multiple of 16); wave id = col tile.
// A layout (16x4 f32): lanes 0-15 rows M=0..15, VGPR0/1 hold K = half*2 +0/+1.
// B layout (4x16 f32): lanes 0-15 cols N=0..15, VGPR0/1 hold K = half*2 +0/+1.
// C/D layout: VGPR r -> M = r + half*8, N = lane&15.
template <int K, int NC>
__global__ __launch_bounds__((NC / 16) * 32)
void gemm_wmma_kernel(const float* __restrict__ X, const float* __restrict__ W,
                      float* __restrict__ Y) {
  const int lane = threadIdx.x & 31;
  const int wave = threadIdx.x >> 5;
  const int half = lane >> 4;
  const int l15  = lane & 15;
  const int m0   = blockIdx.x * 16;
  const int col  = wave * 16 + l15;
  const float* xrow = X + (size_t)(m0 + l15) * K;
  v8f c = {};
#pragma unroll
  for (int k0 = 0; k0 < K; k0 += 4) {
    const int ka = k0 + half * 2;
    const float2 av = *reinterpret_cast<const float2*>(xrow + ka);
    v2f a; a.x = av.x; a.y = av.y;
    v2f b;
    b.x = W[(size_t)ka * NC + col];
    b.y = W[(size_t)(ka + 1) * NC + col];
    c = __builtin_amdgcn_wmma_f32_16x16x4_f32(false, a, false, b, (short)0, c,
                                              false, false);
  }
#pragma unroll
  for (int r = 0; r < 8; ++r)
    Y[(size_t)(m0 + half * 8 + r) * NC + col] = c[r];
}

// --------------------- attention logit precompute -------------------------
// Layer 1: a_src[n,h] = dot(xh1[n, h*16 : h*16+16], att_src1[h]);  H=8, C=16
__global__ void scores1_kernel(const float* __restrict__ xh,
                               const float* __restrict__ ws_,
                               const float* __restrict__ wd_,
                               float* __restrict__ as_, float* __restrict__ ad_,
                               int NH) {
  int t = blockIdx.x * blockDim.x + threadIdx.x;
  if (t >= NH) return;
  int n = t >> 3, h = t & 7;
  const float4* v = reinterpret_cast<const float4*>(xh + (size_t)n * 128 + h * 16);
  const float4* s = reinterpret_cast<const float4*>(ws_ + h * 16);
  const float4* d = reinterpret_cast<const float4*>(wd_ + h * 16);
  float accs = 0.f, accd = 0.f;
#pragma unroll
  for (int q = 0; q < 4; ++q) {
    float4 x = v[q], a = s[q], b = d[q];
    accs += x.x * a.x + x.y * a.y + x.z * a.z + x.w * a.w;
    accd += x.x * b.x + x.y * b.y + x.z * b.z + x.w * b.w;
  }
  as_[t] = accs;
  ad_[t] = accd;
}

// Layer 2: H=1, C=64
__global__ void scores2_kernel(const float* __restrict__ xh,
                               const float* __restrict__ ws_,
                               const float* __restrict__ wd_,
                               float* __restrict__ as_, float* __restrict__ ad_,
                               int N) {
  int n = blockIdx.x * blockDim.x + threadIdx.x;
  if (n >= N) return;
  const float4* v = reinterpret_cast<const float4*>(xh + (size_t)n * 64);
  const float4* s = reinterpret_cast<const float4*>(ws_);
  const float4* d = reinterpret_cast<const float4*>(wd_);
  float accs = 0.f, accd = 0.f;
#pragma unroll
  for (int q = 0; q < 16; ++q) {
    float4 x = v[q], a = s[q], b = d[q];
    accs += x.x * a.x + x.y * a.y + x.z * a.z + x.w * a.w;
    accd += x.x * b.x + x.y * b.y + x.z * b.z + x.w * b.w;
  }
  as_[n] = accs;
  ad_[n] = accd;
}

// --------------------------- edge-phase kernels ---------------------------
template <int H>
__global__ void edge_max_kernel(const int* __restrict__ esrc,
                                const int* __restrict__ edst,
                                const float* __restrict__ as_,
                                const float* __restrict__ ad_,
                                unsigned* __restrict__ m, int E, int total) {
  int t = blockIdx.x * blockDim.x + threadIdx.x;
  if (t >= total) return;
  int e = t / H, h = t - e * H;
  int s, d;
  if (e < E) { s = esrc[e]; d = edst[e]; } else { s = e - E; d = s; }
  float ev = as_[(size_t)s * H + h] + ad_[(size_t)d * H + h];
  ev = ev > 0.f ? ev : NEG_SLOPE * ev;
  atomicMax(&m[(size_t)d * H + h], f2ord(ev));
}

template <int H>
__global__ void edge_denom_kernel(const int* __restrict__ esrc,
                                  const int* __restrict__ edst,
                                  const float* __restrict__ as_,
                                  const float* __restrict__ ad_,
                                  const unsigned* __restrict__ m,
                                  float* __restrict__ denom, int E, int total) {
  int t = blockIdx.x * blockDim.x + threadIdx.x;
  if (t >= total) return;
  int e = t / H, h = t - e * H;
  int s, d;
  if (e < E) { s = esrc[e]; d = edst[e]; } else { s = e - E; d = s; }
  float ev = as_[(size_t)s * H + h] + ad_[(size_t)d * H + h];
  ev = ev > 0.f ? ev : NEG_SLOPE * ev;
  float ex = expf(ev - ord2f(m[(size_t)d * H + h]));
  atomicAdd(&denom[(size_t)d * H + h], ex);
}

// Message scatter-add. Each thread handles CH/NSPLIT channels of one (edge,head).
template <int H, int CH, int NSPLIT>
__global__ void edge_msg_kernel(const int* __restrict__ esrc,
                                const int* __restrict__ edst,
                                const float* __restrict__ as_,
                                const float* __restrict__ ad_,
                                const unsigned* __restrict__ m,
                                const float* __restrict__ denom,
                                const float* __restrict__ xh,
                                float* __restrict__ agg, int E, int total) {
  int t = blockIdx.x * blockDim.x + threadIdx.x;
  if (t >= total) return;
  constexpr int TPE = H * NSPLIT;
  int e = t / TPE;
  int r = t - e * TPE;
  int h = r / NSPLIT;
  int q = r - h * NSPLIT;
  int s, d;
  if (e < E) { s = esrc[e]; d = edst[e]; } else { s = e - E; d = s; }
  float ev = as_[(size_t)s * H + h] + ad_[(size_t)d * H + h];
  ev = ev > 0.f ? ev : NEG_SLOPE * ev;
  float alpha = expf(ev - ord2f(m[(size_t)d * H + h])) /
                (denom[(size_t)d * H + h] + 1e-16f);
  constexpr int F  = H * CH;        // features per node
  constexpr int CW = CH / NSPLIT;   // channels handled by this thread
  const float* xs = xh + (size_t)s * F + h * CH + q * CW;
  float* od = agg + (size_t)d * F + h * CH + q * CW;
#pragma unroll
  for (int c = 0; c < CW; ++c) atomicAdd(&od[c], xs[c] * alpha);
}

// ------------------------------ bias + ELU --------------------------------
__global__ void bias_elu_kernel(float* __restrict__ h, const float* __restrict__ b,
                                int n) {
  int i = blockIdx.x * blockDim.x + threadIdx.x;
  if (i >= n) return;
  float v = h[i] + b[i & 127];
  h[i] = v > 0.f ? v : (expf(v) - 1.f);
}

// ------------------------------ launcher ----------------------------------
extern "C" void kernel_launch(void* const* d_in, const int* in_sizes, int n_in,
                              void* d_out, int out_size, void* d_ws,
                              size_t ws_size, hipStream_t stream) {
  (void)n_in; (void)out_size; (void)ws_size;
  const float* x        = (const float*)d_in[0];
  const int*   eidx     = (const int*)d_in[1];
  const float* W1       = (const float*)d_in[2];
  const float* att_src1 = (const float*)d_in[3];
  const float* att_dst1 = (const float*)d_in[4];
  const float* b1       = (const float*)d_in[5];
  const float* W2       = (const float*)d_in[6];
  const float* att_src2 = (const float*)d_in[7];
  const float* att_dst2 = (const float*)d_in[8];
  const float* b2       = (const float*)d_in[9];
  float* out = (float*)d_out;

  const int N    = in_sizes[0] / 128;  // 50000 (multiple of 16)
  const int E    = in_sizes[1] / 2;    // 800000
  const int Etot = E + N;              // edges + self loops
  const int* esrc = eidx;
  const int* edst = eidx + E;

  // workspace layout (~71 MB)
  float* ws = (float*)d_ws;
  float*    xh1  = ws;            ws += (size_t)N * 128;
  float*    agg1 = ws;            ws += (size_t)N * 128;
  float*    as1  = ws;            ws += (size_t)N * 8;
  float*    ad1  = ws;            ws += (size_t)N * 8;
  unsigned* m1   = (unsigned*)ws; ws += (size_t)N * 8;
  float*    dn1  = ws;            ws += (size_t)N * 8;
  float*    xh2  = ws;            ws += (size_t)N * 64;
  float*    as2  = ws;            ws += (size_t)N;
  float*    ad2  = ws;            ws += (size_t)N;
  unsigned* m2   = (unsigned*)ws; ws += (size_t)N;
  float*    dn2  = ws;

  auto cdiv = [](int a, int b) { return (a + b - 1) / b; };

  // init accumulators (every call: harness does not re-zero)
  fill_u32_kernel<<<cdiv(N * 8, 256), 256, 0, stream>>>(m1, ORD_NEG_INF, N * 8);
  fill_f32_kernel<<<cdiv(N * 8, 256), 256, 0, stream>>>(dn1, 0.f, N * 8);
  fill_f32_kernel<<<cdiv(N * 128, 256), 256, 0, stream>>>(agg1, 0.f, N * 128);
  fill_u32_kernel<<<cdiv(N, 256), 256, 0, stream>>>(m2, ORD_NEG_INF, N);
  fill_f32_kernel<<<cdiv(N, 256), 256, 0, stream>>>(dn2, 0.f, N);
  init_bias_kernel<<<cdiv(N * 64, 256), 256, 0, stream>>>(out, b2, N * 64, 63);

  // ---- layer 1: GAT(128 -> 8 heads x 16), concat ----
  gemm_wmma_kernel<128, 128><<<N / 16, 256, 0, stream>>>(x, W1, xh1);
  scores1_kernel<<<cdiv(N * 8, 256), 256, 0, stream>>>(xh1, att_src1, att_dst1,
                                                       as1, ad1, N * 8);
  edge_max_kernel<8><<<cdiv(Etot * 8, 256), 256, 0, stream>>>(esrc, edst, as1,
                                                              ad1, m1, E, Etot * 8);
  edge_denom_kernel<8><<<cdiv(Etot * 8, 256), 256, 0, stream>>>(esrc, edst, as1,
                                                                ad1, m1, dn1, E, Etot * 8);
  edge_msg_kernel<8, 16, 1><<<cdiv(Etot * 8, 256), 256, 0, stream>>>(
      esrc, edst, as1, ad1, m1, dn1, xh1, agg1, E, Etot * 8);
  bias_elu_kernel<<<cdiv(N * 128, 256), 256, 0, stream>>>(agg1, b1, N * 128);

  // ---- layer 2: GAT(128 -> 1 head x 64), mean(=identity) + bias ----
  gemm_wmma_kernel<128, 64><<<N / 16, 128, 0, stream>>>(agg1, W2, xh2);
  scores2_kernel<<<cdiv(N, 256), 256, 0, stream>>>(xh2, att_src2, att_dst2, as2,
                                                   ad2, N);
  edge_max_kernel<1><<<cdiv(Etot, 256), 256, 0, stream>>>(esrc, edst, as2, ad2,
                                                          m2, E, Etot);
  edge_denom_kernel<1><<<cdiv(Etot, 256), 256, 0, stream>>>(esrc, edst, as2, ad2,
                                                            m2, dn2, E, Etot);
  edge_msg_kernel<1, 64, 4><<<cdiv(Etot * 4, 256), 256, 0, stream>>>(
      esrc, edst, as2, ad2, m2, dn2, xh2, out, E, Etot * 4);
}